// Measure_Possible_Outcome_SI_37709812859568
// MI455X (gfx1250) — compile-verified
//
#include <hip/hip_runtime.h>

// Measure_Possible_Outcome: out[b, v] = sum_k diag(x[b])[k] * P[k, v]
//   x: [4, 8855, 8855] f32, P: [8855, 15] f32 (0/1), out: [4, 15] f32
//
// MI455X strategy: the job is a memory-latency-bound strided gather of
// 35,420 diagonal f32s (each on its own 128B line -> ~4.5 MB of line traffic,
// ~0.2us at 23.3 TB/s). So: one K-chunk of 4 per wave32 wave, branchless body
// (4 clause'd loads in flight before one wait), accumulate with full-precision
// V_WMMA_F32_16X16X4_F32 (M = 4 batches padded to 16, N = 15 outcomes padded
// to 16), LDS-reduce 6 waves/block, atomicAdd the 60 live elements into
// pre-zeroed d_out.
//
// Geometry: nsteps = ceil(8855/4) = 2214 = 6 * 369, so 369 blocks x 6 waves
// gives EXACTLY one K-step per wave -- no guard, no idle waves, EXEC is
// all-1s at every WMMA (required by the ISA; also avoids the exec-masked
// skip path leaving the accumulator VGPRs uninitialized).
//
// Key trick: D[M,N] depends only on A row M and B col N, and we discard
// D rows >= 4 and col 15 -- so out-of-range lanes load *clamped* valid
// addresses (garbage feeds only discarded outputs; all values finite).
// The K tail is handled by zeroing B alone (cndmask, no branch).

typedef __attribute__((ext_vector_type(2))) float v2f;
typedef __attribute__((ext_vector_type(8))) float v8f;

#define DIM    8855
#define NBATCH 4
#define NVAL   15
#define WPB    6     // waves per block (192 threads); 2214 / 6 = 369 blocks

__global__ void zero_out_kernel(float* __restrict__ out) {
    int t = threadIdx.x;
    if (t < NBATCH * NVAL) out[t] = 0.0f;
}

__global__ __launch_bounds__(WPB * 32) void diag_proj_wmma_kernel(
    const float* __restrict__ x,   // [4, DIM, DIM]
    const float* __restrict__ P,   // [DIM, NVAL]
    float* __restrict__ out)       // [4, NVAL], pre-zeroed
{
    const int lane = threadIdx.x & 31;
    const int wib  = threadIdx.x >> 5;          // wave in block (0..5)
    const int wave = blockIdx.x * WPB + wib;    // 0..2213, exactly one step

    // ISA layout, 32-bit 16x4 A-matrix (wave32):
    //   lanes 0-15:  VGPR0 = K=0, VGPR1 = K=1   (M = lane)
    //   lanes 16-31: VGPR0 = K=2, VGPR1 = K=3   (M = lane-16)
    // B (4x16) mirrored: N = lane&15, K pair selected by lane>>4.
    const int half = lane >> 4;
    const int mn   = lane & 15;

    const int k0  = (wave << 2) + (half << 1);  // this lane's K pair base
    const int kc0 = min(k0,     DIM - 1);       // clamped (tail-safe) K
    const int kc1 = min(k0 + 1, DIM - 1);

    // A: diag gather. Lanes mn>=4 clamp to batch 3 (duplicate address,
    // coalesced; result feeds only discarded D rows).
    const float* abase = x + (size_t)min(mn, NBATCH - 1) * DIM * DIM;
    v2f a;
    a.x = abase[(size_t)kc0 * (DIM + 1)];
    a.y = abase[(size_t)kc1 * (DIM + 1)];

    // B: projector columns. Lane mn==15 clamps to col 14 (discarded).
    // Zero B (cndmask) for the K tail so the A*B product vanishes there.
    const int nc = min(mn, NVAL - 1);
    float bx = P[(size_t)kc0 * NVAL + nc];
    float by = P[(size_t)kc1 * NVAL + nc];
    v2f b;
    b.x = (k0     < DIM) ? bx : 0.f;
    b.y = (k0 + 1 < DIM) ? by : 0.f;

    // D = A(16x4) * B(4x16) + 0, full f32 precision. EXEC all-1s by design.
    v8f acc = (v8f){0.f, 0.f, 0.f, 0.f, 0.f, 0.f, 0.f, 0.f};
    acc = __builtin_amdgcn_wmma_f32_16x16x4_f32(
        /*neg_a=*/false, a, /*neg_b=*/false, b,
        /*c_mod=*/(short)0, acc, /*reuse_a=*/false, /*reuse_b=*/false);

    // Only VGPRs 0-3 hold live rows (M = j < 4 on lanes 0-15); reduce those.
    __shared__ float red[WPB][128];
    #pragma unroll
    for (int j = 0; j < 4; ++j) red[wib][j * 32 + lane] = acc[j];
    __syncthreads();

    // Slot t: j = t>>5 (VGPR 0-3), l = t&31 (lane). Live: l<15 on lanes 0-14
    // (M = j < 4, N = l). All live slots have t <= 110 < 128 < blockDim.
    const int t = threadIdx.x;
    if (t < 128) {
        float sum = 0.f;
        #pragma unroll
        for (int w = 0; w < WPB; ++w) sum += red[w][t];
        const int j  = t >> 5;
        const int l  = t & 31;
        const int Mr = j + ((l >> 4) << 3);
        const int Nc = l & 15;
        if (Mr < NBATCH && Nc < NVAL) {
            atomicAdd(&out[Mr * NVAL + Nc], sum);
        }
    }
}

extern "C" void kernel_launch(void* const* d_in, const int* in_sizes, int n_in,
                              void* d_out, int out_size, void* d_ws, size_t ws_size,
                              hipStream_t stream) {
    const float* x = (const float*)d_in[0];  // [4, DIM, DIM]
    const float* P = (const float*)d_in[1];  // [DIM, NVAL]
    float* out = (float*)d_out;              // [4, NVAL]

    zero_out_kernel<<<1, 64, 0, stream>>>(out);

    const int nsteps = (DIM + 3) / 4;        // 2214 = WPB * 369
    const int grid   = nsteps / WPB;         // 369 blocks, zero idle waves
    diag_proj_wmma_kernel<<<grid, WPB * 32, 0, stream>>>(x, P, out);
}